// MultiHeadSelfAttention_5188320494158
// MI455X (gfx1250) — compile-verified
//
#include <hip/hip_runtime.h>
#include <hip/hip_bf16.h>

// MI455X / gfx1250, wave32. All matmuls use V_WMMA_F32_16X16X4_F32 (fp32 in/out,
// matching the fp32 reference). Attention is fused flash-style so the 512MB/layer
// score tensor is never materialized; per-layer working set stays in the 192MB L2.
//
// Workspace layout (floats), total 7 * 4Mi floats = 112 MB:
//   [Q | K | V | CTX | YBUF | HPING0 | HPING1], each B*S*H = 4*2048*512 = 4Mi.

typedef __attribute__((ext_vector_type(2))) float v2f;
typedef __attribute__((ext_vector_type(8))) float v8f;

constexpr int cB  = 4;
constexpr int cS  = 2048;
constexpr int cH  = 512;
constexpr int cNH = 8;
constexpr int cDK = 64;
constexpr int cL  = 4;
constexpr int cM  = cB * cS;          // 8192 rows

constexpr int TM = 128;               // block M tile
constexpr int TN = 64;                // block N tile
constexpr int TK = 16;                // K chunk staged in LDS
constexpr int LSTR = 18;              // LDS row stride (pad: conflict-free, 8B-aligned)

// D(16x16) = A(16x4) * B(4x16) + C, fp32. A: lane&15 -> M row, lane-half -> K pair.
// B: lane&15 -> N col, lane-half -> K pair. C/D: VGPR v -> row v (lo half) / v+8 (hi).
__device__ __forceinline__ v8f wmma_f32(v2f a, v2f b, v8f c) {
  return __builtin_amdgcn_wmma_f32_16x16x4_f32(false, a, false, b, (short)0, c,
                                               false, false);
}

// ---------------------------------------------------------------------------
// Shared WMMA GEMM core: Y[128x64] tile of X(8192x512) @ W(512x512)^T.
// 8 waves; wave w owns rows [w*16, w*16+16) x all 64 cols (4 accumulators).
// ---------------------------------------------------------------------------
__device__ __forceinline__ void gemm_core(const float* __restrict__ X,
                                          const float* __restrict__ W,
                                          float* ldsX, float* ldsW,
                                          int row0blk, int n0blk, v8f acc[4]) {
  const int tid  = threadIdx.x;
  const int lane = tid & 31;
  const int wv   = tid >> 5;          // wave 0..7
  const int lq   = lane & 15;
  const int ko   = (lane >> 4) << 1;  // K-pair offset: 0 (lanes 0-15) / 2 (16-31)
  const int rr   = tid >> 2;          // 0..63
  const int cc   = (tid & 3) << 2;    // 0,4,8,12

  for (int k0 = 0; k0 < cH; k0 += TK) {
    // Stage X[128x16] and W[64x16] (coalesced float4 per thread).
    float4 x0 = *reinterpret_cast<const float4*>(&X[(size_t)(row0blk + rr) * cH + k0 + cc]);
    float4 x1 = *reinterpret_cast<const float4*>(&X[(size_t)(row0blk + rr + 64) * cH + k0 + cc]);
    float4 w0 = *reinterpret_cast<const float4*>(&W[(size_t)(n0blk + rr) * cH + k0 + cc]);
    float* px0 = &ldsX[rr * LSTR + cc];
    px0[0] = x0.x; px0[1] = x0.y; px0[2] = x0.z; px0[3] = x0.w;
    float* px1 = &ldsX[(rr + 64) * LSTR + cc];
    px1[0] = x1.x; px1[1] = x1.y; px1[2] = x1.z; px1[3] = x1.w;
    float* pw0 = &ldsW[rr * LSTR + cc];
    pw0[0] = w0.x; pw0[1] = w0.y; pw0[2] = w0.z; pw0[3] = w0.w;
    __syncthreads();

#pragma unroll
    for (int kk = 0; kk < TK; kk += 4) {
      v2f a = *reinterpret_cast<const v2f*>(&ldsX[(wv * 16 + lq) * LSTR + kk + ko]);
#pragma unroll
      for (int t = 0; t < 4; ++t) {
        v2f b = *reinterpret_cast<const v2f*>(&ldsW[(t * 16 + lq) * LSTR + kk + ko]);
        acc[t] = wmma_f32(a, b, acc[t]);
      }
    }
    __syncthreads();
  }
}

// ---------------------------------------------------------------------------
// Q/K/V projections: blockIdx.z selects {Q,K,V}. Output scattered to (B,NH,S,DK).
// ---------------------------------------------------------------------------
__global__ __launch_bounds__(256) void qkv_gemm_kernel(
    const float* __restrict__ X,
    const float* __restrict__ Wq, const float* __restrict__ bq,
    const float* __restrict__ Wk, const float* __restrict__ bk,
    const float* __restrict__ Wv, const float* __restrict__ bv,
    float* __restrict__ Qo, float* __restrict__ Ko, float* __restrict__ Vo) {
  __shared__ float ldsX[TM * LSTR];
  __shared__ float ldsW[TN * LSTR];
  const float* W; const float* bias; float* out;
  if (blockIdx.z == 0)      { W = Wq; bias = bq; out = Qo; }
  else if (blockIdx.z == 1) { W = Wk; bias = bk; out = Ko; }
  else                      { W = Wv; bias = bv; out = Vo; }

  const int row0blk = blockIdx.y * TM;
  const int n0blk   = blockIdx.x * TN;
  const v8f zero = {0.f, 0.f, 0.f, 0.f, 0.f, 0.f, 0.f, 0.f};
  v8f acc[4] = {zero, zero, zero, zero};

  gemm_core(X, W, ldsX, ldsW, row0blk, n0blk, acc);

  const int lane  = threadIdx.x & 31;
  const int wv    = threadIdx.x >> 5;
  const int lq    = lane & 15;
  const int half8 = (lane >> 4) ? 8 : 0;
  const int row0  = row0blk + wv * 16;
#pragma unroll
  for (int t = 0; t < 4; ++t) {
    const int col = n0blk + t * 16 + lq;
    const int hh = col >> 6, dd = col & 63;   // head, dim
    const float bb = bias[col];
#pragma unroll
    for (int v = 0; v < 8; ++v) {
      const int row  = row0 + v + half8;
      const int bidx = row >> 11, ss = row & (cS - 1);
      out[(((size_t)(bidx * cNH + hh)) * cS + ss) * cDK + dd] = acc[t][v] + bb;
    }
  }
}

// ---------------------------------------------------------------------------
// Fused flash attention. One wave per (b, head, 16-query tile).
// Computes S^T = K·Q^T so the softmax (over keys j) reduces along the VGPR /
// lane-half axis: per-lane math + one shfl_xor(16). O^T = V^T·P^T.
// ---------------------------------------------------------------------------
__global__ __launch_bounds__(32) void attn_kernel(
    const float* __restrict__ Q, const float* __restrict__ K,
    const float* __restrict__ V, const int* __restrict__ mask,
    float* __restrict__ ctx) {
  const int lane  = threadIdx.x & 31;
  const int lq    = lane & 15;
  const int ko    = (lane >> 4) << 1;
  const int half8 = (lane >> 4) ? 8 : 0;
  const bool lo   = lane < 16;
  const int qt = blockIdx.x, head = blockIdx.y, b = blockIdx.z;
  const int q0 = qt * 16;

  const size_t bh = ((size_t)b * cNH + head) * cS * cDK;
  const float* Qb = Q + bh;
  const float* Kb = K + bh;
  const float* Vb = V + bh;
  const int* mrow = mask + ((size_t)b * cS + q0 + lq) * cS;  // row for this lane's q

  // Preload Q^T B-fragments (reused across all 128 key tiles): B[d][q]=Q[q][d].
  v2f qb[16];
#pragma unroll
  for (int i = 0; i < 16; ++i)
    qb[i] = *reinterpret_cast<const v2f*>(&Qb[(q0 + lq) * cDK + i * 4 + ko]);

  const v8f zero = {0.f, 0.f, 0.f, 0.f, 0.f, 0.f, 0.f, 0.f};
  v8f o[4] = {zero, zero, zero, zero};         // O^T accumulators (4 d-tiles)
  float m_i = -3.0e38f, l_i = 0.f;
  const float scale = 0.125f;                  // 1/sqrt(64)

  for (int jt = 0; jt < cS / 16; ++jt) {
    const int jbase = jt * 16;

    // S^T tile = K(16j x 64d) @ Q^T(64d x 16q): 16 chained f32 WMMAs.
    v8f st = zero;
    const float* kp = &Kb[(jbase + lq) * cDK + ko];
#pragma unroll
    for (int i = 0; i < 16; ++i) {
      v2f ka = *reinterpret_cast<const v2f*>(kp + i * 4);
      st = wmma_f32(ka, qb[i], st);
    }

    // Scale + mask, then online softmax over j (VGPR axis + lane-half).
    float p[8];
    float mx = -3.0e38f;
#pragma unroll
    for (int v = 0; v < 8; ++v) {
      const int j = jbase + v + half8;
      const float sv = (mrow[j] == 0) ? -1.0e9f : st[v] * scale;
      p[v] = sv;
      mx = fmaxf(mx, sv);
    }
    mx = fmaxf(mx, __shfl_xor(mx, 16));
    const float mnew = fmaxf(m_i, mx);
    const float corr = __expf(m_i - mnew);     // first iter: exp(-3e38)=0
    float rs = 0.f;
#pragma unroll
    for (int v = 0; v < 8; ++v) { p[v] = __expf(p[v] - mnew); rs += p[v]; }
    rs += __shfl_xor(rs, 16);
    l_i = l_i * corr + rs;
    m_i = mnew;
#pragma unroll
    for (int t = 0; t < 4; ++t)
#pragma unroll
      for (int v = 0; v < 8; ++v) o[t][v] *= corr;

    // Re-layout P^T (C/D layout) into 4 WMMA B-fragments via shfl_xor(16):
    // chunk c covers key rows j0=c*4..c*4+3; lane-half selects the K pair.
    const float x0 = __shfl_xor(p[0], 16), x1 = __shfl_xor(p[1], 16);
    const float x2 = __shfl_xor(p[2], 16), x3 = __shfl_xor(p[3], 16);
    const float x4 = __shfl_xor(p[4], 16), x5 = __shfl_xor(p[5], 16);
    const float x6 = __shfl_xor(p[6], 16), x7 = __shfl_xor(p[7], 16);
    v2f bfr[4];
    bfr[0].x = lo ? p[0] : x2;  bfr[0].y = lo ? p[1] : x3;   // j0 = 0
    bfr[1].x = lo ? p[4] : x6;  bfr[1].y = lo ? p[5] : x7;   // j0 = 4
    bfr[2].x = lo ? x0 : p[2];  bfr[2].y = lo ? x1 : p[3];   // j0 = 8
    bfr[3].x = lo ? x4 : p[6];  bfr[3].y = lo ? x5 : p[7];   // j0 = 12

    // O^T += V^T(16d x 16j) @ P^T(16j x 16q): 16 WMMAs (4 d-tiles x 4 j-chunks).
    const float* vp = Vb + (size_t)jbase * cDK;
#pragma unroll
    for (int dt = 0; dt < 4; ++dt) {
      const int dcol = dt * 16 + lq;
#pragma unroll
      for (int c = 0; c < 4; ++c) {
        const int j0 = c * 4 + ko;
        v2f a;
        a.x = vp[j0 * cDK + dcol];
        a.y = vp[(j0 + 1) * cDK + dcol];
        o[dt] = wmma_f32(a, bfr[c], o[dt]);
      }
    }
  }

  // Normalize and store ctx in (B,S,NH*DK) = (B,S,H) head-concat order.
  const float inv = 1.0f / l_i;
  float* cp = ctx + ((size_t)b * cS + q0 + lq) * cH + head * cDK;
#pragma unroll
  for (int dt = 0; dt < 4; ++dt)
#pragma unroll
    for (int v = 0; v < 8; ++v)
      cp[dt * 16 + v + half8] = o[dt][v] * inv;
}

// ---------------------------------------------------------------------------
// Output projection + residual: Y = ctx @ Wo^T + bo + h_in  (row-major out).
// ---------------------------------------------------------------------------
__global__ __launch_bounds__(256) void oproj_kernel(
    const float* __restrict__ CTX, const float* __restrict__ W,
    const float* __restrict__ bias, const float* __restrict__ Hin,
    float* __restrict__ Yb) {
  __shared__ float ldsX[TM * LSTR];
  __shared__ float ldsW[TN * LSTR];
  const int row0blk = blockIdx.y * TM;
  const int n0blk   = blockIdx.x * TN;
  const v8f zero = {0.f, 0.f, 0.f, 0.f, 0.f, 0.f, 0.f, 0.f};
  v8f acc[4] = {zero, zero, zero, zero};

  gemm_core(CTX, W, ldsX, ldsW, row0blk, n0blk, acc);

  const int lane  = threadIdx.x & 31;
  const int wv    = threadIdx.x >> 5;
  const int lq    = lane & 15;
  const int half8 = (lane >> 4) ? 8 : 0;
  const int row0  = row0blk + wv * 16;
#pragma unroll
  for (int t = 0; t < 4; ++t) {
    const int col = n0blk + t * 16 + lq;
    const float bb = bias[col];
#pragma unroll
    for (int v = 0; v < 8; ++v) {
      const int row = row0 + v + half8;
      const size_t idx = (size_t)row * cH + col;
      Yb[idx] = acc[t][v] + bb + Hin[idx];
    }
  }
}

// ---------------------------------------------------------------------------
// LayerNorm with torch-style unbiased std (ddof=1): one block per row of 512.
// ---------------------------------------------------------------------------
__global__ __launch_bounds__(256) void ln_kernel(
    const float* __restrict__ Y, const float* __restrict__ alpha,
    const float* __restrict__ beta, float* __restrict__ Out) {
  __shared__ float red[16];
  const int r = blockIdx.x;
  const int t = threadIdx.x;
  const float y0 = Y[(size_t)r * cH + t];
  const float y1 = Y[(size_t)r * cH + t + 256];
  float s  = y0 + y1;
  float ss = y0 * y0 + y1 * y1;
#pragma unroll
  for (int off = 16; off > 0; off >>= 1) {
    s  += __shfl_xor(s, off);
    ss += __shfl_xor(ss, off);
  }
  const int lane = t & 31, wv = t >> 5;
  if (lane == 0) { red[wv] = s; red[8 + wv] = ss; }
  __syncthreads();
  float sum = 0.f, sq = 0.f;
#pragma unroll
  for (int i = 0; i < 8; ++i) { sum += red[i]; sq += red[8 + i]; }
  const float mu  = sum * (1.0f / cH);
  const float var = (sq - (float)cH * mu * mu) * (1.0f / (cH - 1));
  const float sd  = sqrtf(fmaxf(var, 0.f));
  const float inv = 1.0f / (sd + 1e-6f);
  Out[(size_t)r * cH + t]       = alpha[t] * (y0 - mu) * inv + beta[t];
  Out[(size_t)r * cH + t + 256] = alpha[t + 256] * (y1 - mu) * inv + beta[t + 256];
}

// ---------------------------------------------------------------------------
extern "C" void kernel_launch(void* const* d_in, const int* in_sizes, int n_in,
                              void* d_out, int out_size, void* d_ws, size_t ws_size,
                              hipStream_t stream) {
  (void)in_sizes; (void)n_in; (void)out_size; (void)ws_size;  // needs >= 112MB ws
  const float* x    = (const float*)d_in[0];
  const int*   mask = (const int*)d_in[1];
  const float* Wq = (const float*)d_in[2];
  const float* bq = (const float*)d_in[3];
  const float* Wk = (const float*)d_in[4];
  const float* bk = (const float*)d_in[5];
  const float* Wv = (const float*)d_in[6];
  const float* bv = (const float*)d_in[7];
  const float* Wo = (const float*)d_in[8];
  const float* bo = (const float*)d_in[9];
  const float* al = (const float*)d_in[10];
  const float* be = (const float*)d_in[11];

  float* ws = (float*)d_ws;
  const size_t NELT = (size_t)cB * cS * cH;   // 4 Mi floats
  float* Qw  = ws;
  float* Kw  = Qw + NELT;
  float* Vw  = Kw + NELT;
  float* CT  = Vw + NELT;
  float* YB  = CT + NELT;
  float* HP0 = YB + NELT;
  float* HP1 = HP0 + NELT;
  float* hp[2] = {HP0, HP1};

  const dim3 gQKV(cH / TN, cM / TM, 3);   // (8, 64, 3)
  const dim3 gO  (cH / TN, cM / TM, 1);
  const dim3 gA  (cS / 16, cNH, cB);      // (128, 8, 4)

  for (int l = 0; l < cL; ++l) {
    const float* hin = (l == 0) ? x : hp[(l + 1) & 1];
    float* hout = (l == cL - 1) ? (float*)d_out : hp[l & 1];
    const size_t wofs = (size_t)l * cH * cH;
    const size_t vofs = (size_t)l * cH;

    hipLaunchKernelGGL(qkv_gemm_kernel, gQKV, dim3(256), 0, stream,
                       hin, Wq + wofs, bq + vofs, Wk + wofs, bk + vofs,
                       Wv + wofs, bv + vofs, Qw, Kw, Vw);
    hipLaunchKernelGGL(attn_kernel, gA, dim3(32), 0, stream,
                       Qw, Kw, Vw, mask, CT);
    hipLaunchKernelGGL(oproj_kernel, gO, dim3(256), 0, stream,
                       CT, Wo + wofs, bo + vofs, hin, YB);
    hipLaunchKernelGGL(ln_kernel, dim3(cM), dim3(256), 0, stream,
                       YB, al + vofs, be + vofs, hout);
  }
}